// MultiUserModel_60361470378694
// MI455X (gfx1250) — compile-verified
//
#include <hip/hip_runtime.h>

#define H 128
#define EPS 1e-5f
#define TP 132               // padded LDS row stride in floats (bank-conflict-free A reads)
#define ROWS_PER_BLOCK 64
#define THREADS 128          // 4 wave32 waves; each wave owns 16 rows

typedef float f2 __attribute__((ext_vector_type(2)));
typedef float f4 __attribute__((ext_vector_type(4)));
typedef float f8 __attribute__((ext_vector_type(8)));

// wc[o*H + h] = (w1[o,h] + w2[o,h]) * weight[h]
// (weight depends on h == WMMA K dimension, so it folds into the B matrix)
__global__ void wc_prep_kernel(const float* __restrict__ w1,
                               const float* __restrict__ w2,
                               const float* __restrict__ weight,
                               float* __restrict__ wc) {
    int i = blockIdx.x * blockDim.x + threadIdx.x;   // 0..H*H-1
    wc[i] = (w1[i] + w2[i]) * weight[i & (H - 1)];
}

__global__ __launch_bounds__(THREADS)
void fused_rmsnorm_gemm_kernel(const float* __restrict__ residual,
                               const float* __restrict__ attn,
                               const float* __restrict__ moe,
                               const float* __restrict__ wc,
                               float* __restrict__ out_combined,
                               float* __restrict__ out_nextres) {
    __shared__ float sTile[ROWS_PER_BLOCK * TP];   // add tile, padded rows
    __shared__ float sSum[ROWS_PER_BLOCK];         // per-row sum of squares

    const int t = threadIdx.x;
    const size_t R0 = (size_t)blockIdx.x * ROWS_PER_BLOCK;

    // ---------------- Phase 1: elementwise pass (coalesced float4) -------------
    // Each iteration, one wave covers exactly one full row (32 lanes x float4 = 128).
    #pragma unroll
    for (int i = 0; i < 16; ++i) {
        int flat4 = i * THREADS + t;          // 0..2047 float4 slots
        int row   = flat4 >> 5;               // 0..63
        int c4    = flat4 & 31;               // float4 column
        size_t g  = (R0 + (size_t)row) * H + (size_t)c4 * 4;

        f4 rv = *(const f4*)(residual + g);
        f4 av = *(const f4*)(attn + g);
        f4 mv = *(const f4*)(moe + g);
        f4 add = rv + av;

        *(f4*)&sTile[row * TP + c4 * 4] = add;     // stage for WMMA A operand
        *(f4*)(out_nextres + g) = add + mv;        // next_residual out

        // wave32 reduction of sum-of-squares for this row
        float ss = add.x*add.x + add.y*add.y + add.z*add.z + add.w*add.w;
        ss += __shfl_xor(ss, 16, 32);
        ss += __shfl_xor(ss, 8, 32);
        ss += __shfl_xor(ss, 4, 32);
        ss += __shfl_xor(ss, 2, 32);
        ss += __shfl_xor(ss, 1, 32);
        if ((t & 31) == 0) sSum[row] = ss;
    }
    __syncthreads();

    // ---------------- Phase 2: WMMA f32 16x16x4, K-loop over 128 ---------------
    const int w      = t >> 5;        // wave id 0..3
    const int lane   = t & 31;
    const int lrow   = lane & 15;     // M within 16-row tile / N within 16-col tile
    const int khalf  = lane >> 4;     // lanes 16..31 hold K+2 per ISA A/B layout
    const int rowloc = w * 16;        // wave's local row base

    // rsqrt scale per C/D-layout row: M = v + 8*khalf
    float s[8];
    #pragma unroll
    for (int v = 0; v < 8; ++v)
        s[v] = rsqrtf(sSum[rowloc + v + 8 * khalf] * (1.0f / H) + EPS);

    // A fragments: lane holds row (rowloc+lrow), K = 4*kk + 2*khalf + {0,1}
    f2 a[32];
    #pragma unroll
    for (int kk = 0; kk < 32; ++kk) {
        int k = kk * 4 + 2 * khalf;
        a[kk] = *(const f2*)&sTile[(rowloc + lrow) * TP + k];
    }

    // 8 output tiles of 16x16 along O; B[k][n] = wc[n][k] (wc is L2-resident, 64KB)
    for (int ot = 0; ot < 8; ++ot) {
        int n = ot * 16 + lrow;
        f8 c = {};
        #pragma unroll
        for (int kk = 0; kk < 32; ++kk) {
            int k = kk * 4 + 2 * khalf;
            f2 b = *(const f2*)(wc + n * H + k);
            c = __builtin_amdgcn_wmma_f32_16x16x4_f32(
                    /*neg_a=*/false, a[kk], /*neg_b=*/false, b,
                    /*c_mod=*/(short)0, c, /*reuse_a=*/false, /*reuse_b=*/false);
        }
        // D layout: VGPR v, lane -> M = v + 8*khalf, N = lrow
        #pragma unroll
        for (int v = 0; v < 8; ++v) {
            size_t grow = R0 + (size_t)(rowloc + v + 8 * khalf);
            out_combined[grow * H + n] = c[v] * s[v];
        }
    }
}

extern "C" void kernel_launch(void* const* d_in, const int* in_sizes, int n_in,
                              void* d_out, int out_size, void* d_ws, size_t ws_size,
                              hipStream_t stream) {
    const float* residual = (const float*)d_in[0];
    const float* attn     = (const float*)d_in[1];
    const float* moe      = (const float*)d_in[2];
    const float* weight   = (const float*)d_in[3];
    const float* w1       = (const float*)d_in[4];
    const float* w2       = (const float*)d_in[5];

    float* out = (float*)d_out;                 // [combined | next_residual]
    float* wc  = (float*)d_ws;                  // 64 KB scratch: (w1+w2)*weight

    const long long nRows = (long long)in_sizes[0] / H;   // 64*4096 = 262144

    wc_prep_kernel<<<(H * H) / 256, 256, 0, stream>>>(w1, w2, weight, wc);

    fused_rmsnorm_gemm_kernel<<<(int)(nRows / ROWS_PER_BLOCK), THREADS, 0, stream>>>(
        residual, attn, moe, wc, out, out + (size_t)nRows * H);
}